// NTM_43301860278667
// MI455X (gfx1250) — compile-verified
//
#include <hip/hip_runtime.h>
#include <hip/hip_bf16.h>
#include <math.h>

#define B_   64
#define N_   8192
#define D_   128
#define H_   512
#define NH_  4
#define GDR_ 134
#define GDW_ 390
#define KG_  1152   // 640 (ctrl) + 512 (h0)
#define EPS_ 1e-10f

typedef float v2f __attribute__((ext_vector_type(2)));
typedef float v4f __attribute__((ext_vector_type(4)));
typedef float v8f __attribute__((ext_vector_type(8)));

__device__ __forceinline__ float sigmoidf_(float x){ return 1.0f/(1.0f+__expf(-x)); }
__device__ __forceinline__ float softplusf_(float x){ return (x>20.0f)? x : log1pf(__expf(x)); }
__device__ __forceinline__ float wred_(float v){
#pragma unroll
  for (int o=16;o;o>>=1) v += __shfl_xor(v,o,32);
  return v;
}
// non-temporal 128-bit load: big streaming tensors must not thrash the 192MB L2
__device__ __forceinline__ v4f ldnt4_(const float* p){
  return __builtin_nontemporal_load((const v4f*)p);
}
__device__ __forceinline__ v4f ld4_(const float* p){ return *(const v4f*)p; }

// ---------------- concat: A_cat = [x | prev_y | prev_read | h0]  [64,1152]
__global__ void k_concat(const float* __restrict__ x, const float* __restrict__ py,
                         const float* __restrict__ pr, const float* __restrict__ h0,
                         float* __restrict__ A){
  int idx = blockIdx.x*blockDim.x + threadIdx.x;
  if (idx >= B_*KG_) return;
  int b = idx / KG_, j = idx % KG_;
  float v;
  if (j < 64)        v = x [b*64  + j];
  else if (j < 128)  v = py[b*64  + (j-64)];
  else if (j < 640)  v = pr[b*512 + (j-128)];
  else               v = h0[b*512 + (j-640)];
  A[idx] = v;
}

// ---------------- gates GEMM: gates[64,2048] = Acat @ [W_ih|W_hh]^T + b_ih + b_hh
// one wave computes a 16x16 tile via V_WMMA_F32_16X16X4_F32
__global__ void k_gemm_gates(const float* __restrict__ A, const float* __restrict__ Wih,
                             const float* __restrict__ Whh, const float* __restrict__ bih,
                             const float* __restrict__ bhh, float* __restrict__ out){
  int lane = threadIdx.x & 31;
  int wv   = threadIdx.x >> 5;
  int tileN = (blockIdx.x*4 + wv)*16;
  int tileM = blockIdx.y*16;
  int q    = lane & 15;
  int half = lane >> 4;
  int nc   = tileN + q;
  v8f acc = {0.f,0.f,0.f,0.f,0.f,0.f,0.f,0.f};
  const float* arow = A + (size_t)(tileM + q)*KG_;
  for (int k0=0; k0<KG_; k0+=4){
    int ka = k0 + 2*half;
    v2f a; a[0] = arow[ka]; a[1] = arow[ka+1];
    v2f bb;
    if (ka < 640){ const float* w = Wih + (size_t)nc*640 + ka;       bb[0]=w[0]; bb[1]=w[1]; }
    else         { const float* w = Whh + (size_t)nc*512 + (ka-640); bb[0]=w[0]; bb[1]=w[1]; }
    acc = __builtin_amdgcn_wmma_f32_16x16x4_f32(false, a, false, bb, (short)0, acc, false, false);
  }
  float bias = bih[nc] + bhh[nc];
#pragma unroll
  for (int j=0;j<8;j++){
    int row = tileM + j + 8*half;
    out[(size_t)row*2048 + nc] = acc[j] + bias;
  }
}

// ---------------- head GEMM: out[64,Ncols] = h @ W^T + b   (K = 512)
__global__ void k_gemm_head(const float* __restrict__ Hm, const float* __restrict__ W,
                            const float* __restrict__ bias, float* __restrict__ out, int Ncols){
  int lane = threadIdx.x & 31;
  int wv   = threadIdx.x >> 5;
  int tileN = (blockIdx.x*4 + wv)*16;
  if (tileN >= Ncols) return;
  int tileM = blockIdx.y*16;
  int q = lane & 15, half = lane>>4;
  int col = tileN + q;
  int nc  = (col < Ncols) ? col : (Ncols-1);
  v8f acc = {0.f,0.f,0.f,0.f,0.f,0.f,0.f,0.f};
  const float* arow = Hm + (size_t)(tileM+q)*512;
  const float* wrow = W  + (size_t)nc*512;
  for (int k0=0;k0<512;k0+=4){
    int ka = k0 + 2*half;
    v2f a;  a[0]=arow[ka];  a[1]=arow[ka+1];
    v2f bb; bb[0]=wrow[ka]; bb[1]=wrow[ka+1];
    acc = __builtin_amdgcn_wmma_f32_16x16x4_f32(false, a, false, bb, (short)0, acc, false, false);
  }
  if (col < Ncols){
    float bs = bias[col];
#pragma unroll
    for (int j=0;j<8;j++)
      out[(size_t)(tileM + j + 8*half)*Ncols + col] = acc[j] + bs;
  }
}

// ---------------- LSTM cell
__global__ void k_lstm(const float* __restrict__ gates, const float* __restrict__ c0,
                       float* __restrict__ hout, float* __restrict__ cout){
  int idx = blockIdx.x*blockDim.x + threadIdx.x;
  if (idx >= B_*H_) return;
  int b = idx/512, j = idx%512;
  const float* g = gates + (size_t)b*2048;
  float i = g[j], f = g[512+j], gg = g[1024+j], o = g[1536+j];
  float c = sigmoidf_(f)*c0[idx] + sigmoidf_(i)*tanhf(gg);
  float h = sigmoidf_(o)*tanhf(c);
  hout[idx]=h; cout[idx]=c;
}

// ---------------- head param prep: normalize key, compute scalars, copy erase/add
__global__ void k_head_params(const float* __restrict__ o, int gd, int has_ea,
                              float* __restrict__ kn, float* __restrict__ scal,
                              float* __restrict__ er, float* __restrict__ ad){
  __shared__ float red[128];
  int p = blockIdx.x;             // b*NH + h
  int t = threadIdx.x;            // 0..127
  const float* op = o + (size_t)p*gd;
  float kd = op[t];
  red[t] = kd*kd;
  __syncthreads();
  for (int s=64;s;s>>=1){ if (t<s) red[t]+=red[t+s]; __syncthreads(); }
  float inv = 1.0f/(sqrtf(red[0])+EPS_);
  kn[(size_t)p*128 + t] = kd*inv;
  if (has_ea){
    er[(size_t)p*128 + t] = op[D_+6     + t];
    ad[(size_t)p*128 + t] = op[2*D_+6   + t];
  }
  if (t==0){
    float beta  = softplusf_(op[D_]);
    float g     = sigmoidf_ (op[D_+1]);
    float a0 = softplusf_(op[D_+2]), a1 = softplusf_(op[D_+3]), a2 = softplusf_(op[D_+4]);
    float mx = fmaxf(a0, fmaxf(a1,a2));
    float e0=__expf(a0-mx), e1=__expf(a1-mx), e2=__expf(a2-mx);
    float ssm = e0+e1+e2;
    float gamma = 1.0f + softplusf_(op[D_+5]);
    float* sc = scal + p*6;
    sc[0]=beta; sc[1]=g; sc[2]=e0/ssm; sc[3]=e1/ssm; sc[4]=e2/ssm; sc[5]=gamma;
  }
}

// ---------------- write-head content similarity over original memory (stream pass 1)
__global__ void k_simw(const float* __restrict__ mem, const float* __restrict__ knw,
                       float* __restrict__ sim){
  int b = blockIdx.y;
  int n = blockIdx.x*8 + (threadIdx.x>>5);
  int lane = threadIdx.x & 31;
  size_t pb = (size_t)b*4;
  v4f k0 = ld4_(knw + (pb+0)*128 + lane*4);
  v4f k1 = ld4_(knw + (pb+1)*128 + lane*4);
  v4f k2 = ld4_(knw + (pb+2)*128 + lane*4);
  v4f k3 = ld4_(knw + (pb+3)*128 + lane*4);
  v4f m = ldnt4_(mem + ((size_t)b*N_ + n)*D_ + lane*4);   // streaming, no L2 retention
  float ss = wred_(m.x*m.x + m.y*m.y + m.z*m.z + m.w*m.w);
  float inv = 1.0f/(sqrtf(ss)+EPS_);
  float d0 = wred_(k0.x*m.x + k0.y*m.y + k0.z*m.z + k0.w*m.w);
  float d1 = wred_(k1.x*m.x + k1.y*m.y + k1.z*m.z + k1.w*m.w);
  float d2 = wred_(k2.x*m.x + k2.y*m.y + k2.z*m.z + k2.w*m.w);
  float d3 = wred_(k3.x*m.x + k3.y*m.y + k3.z*m.z + k3.w*m.w);
  if (lane==0){
    size_t base = pb*N_ + n;
    sim[base        ] = d0*inv;
    sim[base +   N_ ] = d1*inv;
    sim[base + 2*N_ ] = d2*inv;
    sim[base + 3*N_ ] = d3*inv;
  }
}

// ---------------- fused: memory update (4 write heads, sequential) + read-head sim (stream pass 2)
__global__ void k_memupd(const float* __restrict__ mem_in, const float* __restrict__ ww,
                         const float* __restrict__ er, const float* __restrict__ ad,
                         const float* __restrict__ knr, float* __restrict__ mem_out,
                         float* __restrict__ simr){
  int b = blockIdx.y;
  int n = blockIdx.x*8 + (threadIdx.x>>5);
  int lane = threadIdx.x & 31;
  size_t pb = (size_t)b*4;
  v4f m = ldnt4_(mem_in + ((size_t)b*N_ + n)*D_ + lane*4);  // last use of original memory
#pragma unroll
  for (int h=0; h<4; h++){
    float wv = ww[(pb+h)*N_ + n];
    v4f e = ld4_(er + (pb+h)*128 + lane*4);
    v4f a = ld4_(ad + (pb+h)*128 + lane*4);
    m.x = m.x*(1.0f - wv*e.x) + wv*a.x;
    m.y = m.y*(1.0f - wv*e.y) + wv*a.y;
    m.z = m.z*(1.0f - wv*e.z) + wv*a.z;
    m.w = m.w*(1.0f - wv*e.w) + wv*a.w;
  }
  *(v4f*)(mem_out + ((size_t)b*N_ + n)*D_ + lane*4) = m;    // RT store: tail stays in L2 for k_read
  float ss = wred_(m.x*m.x + m.y*m.y + m.z*m.z + m.w*m.w);
  float inv = 1.0f/(sqrtf(ss)+EPS_);
#pragma unroll
  for (int h=0; h<4; h++){
    v4f k = ld4_(knr + (pb+h)*128 + lane*4);
    float d = wred_(k.x*m.x + k.y*m.y + k.z*m.z + k.w*m.w);
    if (lane==0) simr[(pb+h)*N_ + n] = d*inv;
  }
}

// ---------------- NTM addressing: softmax -> interpolate -> shift -> sharpen -> normalize
__global__ void k_address(float* __restrict__ sim, const float* __restrict__ scal,
                          const float* __restrict__ prevw, float* __restrict__ wg,
                          float* __restrict__ wout){
  __shared__ float red[256];
  int p = blockIdx.x, t = threadIdx.x;
  const float* sc = scal + p*6;
  float beta=sc[0], g=sc[1], s0=sc[2], s1=sc[3], s2=sc[4], gamma=sc[5];
  float* sp  = sim   + (size_t)p*N_;
  float* wgp = wg    + (size_t)p*N_;
  const float* pw = prevw + (size_t)p*N_;
  float* wo  = wout  + (size_t)p*N_;
  // pass 1: max(beta*sim)
  float mv = -3.0e38f;
  for (int n=t; n<N_; n+=256) mv = fmaxf(mv, beta*sp[n]);
  red[t]=mv; __syncthreads();
  for (int s=128;s;s>>=1){ if(t<s) red[t]=fmaxf(red[t],red[t+s]); __syncthreads(); }
  float smax = red[0]; __syncthreads();
  // pass 2: exp + sum
  float sum=0.f;
  for (int n=t;n<N_;n+=256){ float e=__expf(beta*sp[n]-smax); wgp[n]=e; sum+=e; }
  red[t]=sum; __syncthreads();
  for (int s=128;s;s>>=1){ if(t<s) red[t]+=red[t+s]; __syncthreads(); }
  float rs = 1.0f/red[0]; __syncthreads();
  // pass 3: interpolate with prev weights
  for (int n=t;n<N_;n+=256) wgp[n] = g*(wgp[n]*rs) + (1.0f-g)*pw[n];
  __syncthreads();
  // pass 4: circular shift conv + sharpen (gamma power), accumulate sum
  float sum2=0.f;
  for (int n=t;n<N_;n+=256){
    float wl = wgp[(n + N_ - 1) & (N_-1)];
    float wr = wgp[(n + 1)      & (N_-1)];
    float wsv = s0*wl + s1*wgp[n] + s2*wr;
    float wp = powf(wsv, gamma);
    sp[n] = wp; sum2 += wp;
  }
  red[t]=sum2; __syncthreads();
  for (int s=128;s;s>>=1){ if(t<s) red[t]+=red[t+s]; __syncthreads(); }
  float invs = 1.0f/(red[0] + EPS_);
  // pass 5: normalize -> final weights
  for (int n=t;n<N_;n+=256) wo[n] = sp[n]*invs;
}

__global__ void k_zero(float* __restrict__ p, int n){
  int i = blockIdx.x*blockDim.x + threadIdx.x;
  if (i < n) p[i] = 0.f;
}

// ---------------- weighted read r[b,h,:] = sum_n w_r[b,h,n] * mem[b,n,:]  (stream pass 3)
// traversed in REVERSE so the first-scheduled blocks consume the most-recently-written
// (still L2-resident) tail of mem_out; loads are non-temporal (last use).
__global__ void k_read(const float* __restrict__ mem, const float* __restrict__ wr,
                       float* __restrict__ r){
  __shared__ float lw[4*512];
  int b  = (B_-1) - blockIdx.y;
  int ch = 15     - blockIdx.x;
  int t  = threadIdx.x;                                   // 128 threads, lane = d
  for (int i=t; i<2048; i+=128){
    int hh = i >> 9, nn = i & 511;
    lw[i] = wr[((size_t)b*4+hh)*N_ + (size_t)ch*512 + nn];
  }
  __syncthreads();
  float a0=0.f,a1=0.f,a2=0.f,a3=0.f;
  const float* mrow = mem + ((size_t)b*N_ + (size_t)ch*512)*D_ + t;
  for (int nn=0; nn<512; nn++){
    float m = __builtin_nontemporal_load(mrow + (size_t)nn*D_);
    a0 += lw[nn]*m; a1 += lw[512+nn]*m; a2 += lw[1024+nn]*m; a3 += lw[1536+nn]*m;
  }
  atomicAdd(&r[((size_t)b*4+0)*128 + t], a0);
  atomicAdd(&r[((size_t)b*4+1)*128 + t], a1);
  atomicAdd(&r[((size_t)b*4+2)*128 + t], a2);
  atomicAdd(&r[((size_t)b*4+3)*128 + t], a3);
}

// ---------------- output GEMM: y[64,64] = tanh([h|r] @ W_out^T + b_out), K = 1024
__global__ void k_gemm_out(const float* __restrict__ Hm, const float* __restrict__ R,
                           const float* __restrict__ W, const float* __restrict__ bias,
                           float* __restrict__ y){
  int lane = threadIdx.x & 31;
  int wv   = threadIdx.x >> 5;
  int tileN = wv*16;
  int tileM = blockIdx.y*16;
  int q = lane & 15, half = lane>>4;
  int col = tileN + q;
  v8f acc = {0.f,0.f,0.f,0.f,0.f,0.f,0.f,0.f};
  const float* hrow = Hm + (size_t)(tileM+q)*512;
  const float* rrow = R  + (size_t)(tileM+q)*512;
  const float* wrow = W  + (size_t)col*1024;
  for (int k0=0;k0<1024;k0+=4){
    int ka = k0 + 2*half;
    v2f a;
    if (ka < 512){ a[0]=hrow[ka];     a[1]=hrow[ka+1];     }
    else         { a[0]=rrow[ka-512]; a[1]=rrow[ka-511];   }
    v2f bb; bb[0]=wrow[ka]; bb[1]=wrow[ka+1];
    acc = __builtin_amdgcn_wmma_f32_16x16x4_f32(false, a, false, bb, (short)0, acc, false, false);
  }
  float bs = bias[col];
#pragma unroll
  for (int j=0;j<8;j++)
    y[(size_t)(tileM + j + 8*half)*64 + col] = tanhf(acc[j] + bs);
}

extern "C" void kernel_launch(void* const* d_in, const int* in_sizes, int n_in,
                              void* d_out, int out_size, void* d_ws, size_t ws_size,
                              hipStream_t stream){
  const float* x    = (const float*)d_in[0];
  const float* py   = (const float*)d_in[1];
  const float* pr   = (const float*)d_in[2];
  const float* mem  = (const float*)d_in[3];
  const float* pwr  = (const float*)d_in[4];
  const float* pww  = (const float*)d_in[5];
  const float* h0   = (const float*)d_in[6];
  const float* c0   = (const float*)d_in[7];
  const float* Wih  = (const float*)d_in[8];
  const float* Whh  = (const float*)d_in[9];
  const float* bih  = (const float*)d_in[10];
  const float* bhh  = (const float*)d_in[11];
  const float* Wrh  = (const float*)d_in[12];
  const float* brh  = (const float*)d_in[13];
  const float* Wwh  = (const float*)d_in[14];
  const float* bwh  = (const float*)d_in[15];
  const float* Wout = (const float*)d_in[16];
  const float* bout = (const float*)d_in[17];

  float* out   = (float*)d_out;
  float* o_y   = out;                                  // [64,64]
  float* o_mem = out + 4096;                           // [64,8192,128]
  float* o_h   = o_mem + (size_t)B_*N_*D_;             // [64,512]
  float* o_c   = o_h + B_*H_;                          // [64,512]
  float* o_wr  = o_c + B_*H_;                          // [64,4,8192]
  float* o_ww  = o_wr + (size_t)B_*NH_*N_;             // [64,4,8192]

  float* ws    = (float*)d_ws;
  float* Acat  = ws;  ws += B_*KG_;                    // 73728
  float* gates = ws;  ws += B_*2048;                   // 131072
  float* ow    = ws;  ws += B_*NH_*GDW_;               // 99840
  float* orh   = ws;  ws += B_*NH_*GDR_;               // 34304
  float* knw   = ws;  ws += 256*128;
  float* knr   = ws;  ws += 256*128;
  float* er    = ws;  ws += 256*128;
  float* ad    = ws;  ws += 256*128;
  float* scw   = ws;  ws += 256*8;
  float* scr   = ws;  ws += 256*8;
  float* simw  = ws;  ws += (size_t)256*N_;            // 2M floats (reused as sim_r)
  float* wg    = ws;  ws += (size_t)256*N_;            // 2M floats
  float* rbuf  = ws;  ws += B_*512;
  float* simr  = simw;   // simw fully consumed before k_memupd writes sim_r

  k_concat      <<<dim3((B_*KG_+255)/256), 256, 0, stream>>>(x, py, pr, h0, Acat);
  k_gemm_gates  <<<dim3(32,4), 128, 0, stream>>>(Acat, Wih, Whh, bih, bhh, gates);
  k_lstm        <<<dim3(128), 256, 0, stream>>>(gates, c0, o_h, o_c);
  k_gemm_head   <<<dim3(25,4), 128, 0, stream>>>(o_h, Wwh, bwh, ow,  NH_*GDW_);
  k_gemm_head   <<<dim3(9,4),  128, 0, stream>>>(o_h, Wrh, brh, orh, NH_*GDR_);
  k_head_params <<<dim3(256), 128, 0, stream>>>(ow,  GDW_, 1, knw, scw, er, ad);
  k_head_params <<<dim3(256), 128, 0, stream>>>(orh, GDR_, 0, knr, scr, nullptr, nullptr);
  k_simw        <<<dim3(N_/8, B_), 256, 0, stream>>>(mem, knw, simw);
  k_address     <<<dim3(256), 256, 0, stream>>>(simw, scw, pww, wg, o_ww);
  k_memupd      <<<dim3(N_/8, B_), 256, 0, stream>>>(mem, o_ww, er, ad, knr, o_mem, simr);
  k_address     <<<dim3(256), 256, 0, stream>>>(simr, scr, pwr, wg, o_wr);
  k_zero        <<<dim3(128), 256, 0, stream>>>(rbuf, B_*512);
  k_read        <<<dim3(16, B_), 128, 0, stream>>>(o_mem, o_wr, rbuf);
  k_gemm_out    <<<dim3(1,4), 128, 0, stream>>>(o_h, rbuf, Wout, bout, o_y);
}